// RUNG_confidence_lambda_65412351918231
// MI455X (gfx1250) — compile-verified
//
#include <hip/hip_runtime.h>
#include <hip/hip_bf16.h>
#include <math.h>
#include <stdint.h>

typedef __attribute__((ext_vector_type(2))) float v2f;
typedef __attribute__((ext_vector_type(8))) float v8f;

#define NN 4096
#define DIN 512
#define HH 256
#define CC 64
#define KSTEPS 10
#define SLICES 8                 // J-dimension split for occupancy
#define JPS ((NN / 16) / SLICES) // J tiles per slice = 32

__device__ __forceinline__ constexpr float kScadA()   { return 3.7f; }
__device__ __forceinline__ constexpr float kLamBase() { return 1.0f/0.9f - 1.0f; }
__device__ __forceinline__ constexpr float kEps()     { return 1e-6f; }
__device__ __forceinline__ constexpr float kDistEps() { return 1e-8f; }

// Low 32 bits of a generic pointer to LDS == wave-relative LDS byte offset.
__device__ __forceinline__ uint32_t lds_off(const void* p) {
  return (uint32_t)(uintptr_t)p;
}

// ---------------------------------------------------------------------------
// init: zero the raw_lam sum accumulator
// ---------------------------------------------------------------------------
__global__ void init_kernel(float* accum) {
  if (threadIdx.x == 0 && blockIdx.x == 0) accum[0] = 0.0f;
}

// ---------------------------------------------------------------------------
// Generic fp32 WMMA GEMM: Cout[M x Nn] = act(Amat[M x Kk] @ B[Kk x Nn] + bias)
// ---------------------------------------------------------------------------
template <bool RELU>
__global__ __launch_bounds__(256) void gemm_bias_kernel(
    const float* __restrict__ Amat, const float* __restrict__ Bmat,
    const float* __restrict__ bias, float* __restrict__ Cout,
    int M, int Nn, int Kk) {
  const int lane = threadIdx.x & 31;
  const int li   = lane & 15;
  const int hf   = lane >> 4;
  const int wave = threadIdx.x >> 5;
  const int tiles_n = Nn >> 4;
  const int t  = blockIdx.x * 8 + wave;
  const int tm = t / tiles_n;
  const int tn = t % tiles_n;

  v8f acc = {0.f,0.f,0.f,0.f,0.f,0.f,0.f,0.f};
  const float* Ar = Amat + (size_t)(tm * 16 + li) * Kk;
  for (int k0 = 0; k0 < Kk; k0 += 4) {
    v2f a;
    a.x = Ar[k0 + 2 * hf];
    a.y = Ar[k0 + 2 * hf + 1];
    v2f b;
    const int kb = k0 + 2 * hf;
    b.x = Bmat[(size_t)kb * Nn + tn * 16 + li];
    b.y = Bmat[(size_t)(kb + 1) * Nn + tn * 16 + li];
    acc = __builtin_amdgcn_wmma_f32_16x16x4_f32(false, a, false, b, (short)0,
                                                acc, false, false);
  }
  const int col = tn * 16 + li;
  const float bv = bias[col];
#pragma unroll
  for (int r = 0; r < 8; ++r) {
    const int row = tm * 16 + r + 8 * hf;
    float v = acc[r] + bv;
    if (RELU) v = fmaxf(v, 0.0f);
    Cout[(size_t)row * Nn + col] = v;
  }
}

// ---------------------------------------------------------------------------
// Per-node confidence -> raw_lam, plus global sum (for the mean)
// ---------------------------------------------------------------------------
__global__ __launch_bounds__(256) void conf_kernel(
    const float* __restrict__ F0, const float* __restrict__ raw_alpha_p,
    float* __restrict__ raw_lam, float* __restrict__ accum) {
  __shared__ float sred[256];
  const int tid = threadIdx.x;
  const int i = blockIdx.x * 256 + tid;
  const float* row = F0 + (size_t)i * CC;
  float m = -1e30f;
#pragma unroll 4
  for (int c = 0; c < CC; ++c) m = fmaxf(m, row[c]);
  float S = 0.0f;
#pragma unroll 4
  for (int c = 0; c < CC; ++c) S += __expf(row[c] - m);
  float conf = fminf(fmaxf(1.0f / S, kEps()), 1.0f - kEps());
  const float alpha = log1pf(__expf(raw_alpha_p[0])) + 0.5f;
  float rl = kLamBase() * powf(fmaxf(1.0f - conf, kEps()), alpha);
  raw_lam[i] = rl;
  sred[tid] = rl;
  __syncthreads();
  for (int s = 128; s > 0; s >>= 1) {
    if (tid < s) sred[tid] += sred[tid + s];
    __syncthreads();
  }
  if (tid == 0) atomicAdd(accum, sred[0]);
}

__global__ __launch_bounds__(256) void lamnode_kernel(
    const float* __restrict__ raw_lam, const float* __restrict__ accum,
    float* __restrict__ lam_node) {
  const int i = blockIdx.x * 256 + threadIdx.x;
  const float mean = accum[0] * (1.0f / (float)NN);
  const float scale = kLamBase() / fmaxf(mean, kEps());
  lam_node[i] = fmaxf(raw_lam[i] * scale, kEps());
}

// ---------------------------------------------------------------------------
// Row degrees of (A + I) -> rinv = 1/sqrt(deg). One block per row.
// ---------------------------------------------------------------------------
__global__ __launch_bounds__(256) void deg_kernel(const float* __restrict__ Adj,
                                                  float* __restrict__ rinv) {
  __shared__ float sred[256];
  const int row = blockIdx.x;
  const int tid = threadIdx.x;
  const float4* r4 = (const float4*)(Adj + (size_t)row * NN);
  float s = 0.0f;
  for (int j = tid; j < NN / 4; j += 256) {
    float4 v = r4[j];
    s += v.x + v.y + v.z + v.w;
  }
  sred[tid] = s;
  __syncthreads();
  for (int s2 = 128; s2 > 0; s2 >>= 1) {
    if (tid < s2) sred[tid] += sred[tid + s2];
    __syncthreads();
  }
  if (tid == 0) rinv[row] = rsqrtf(sred[0] + 1.0f);
}

// ---------------------------------------------------------------------------
// Y = F * rinv (row-scaled), sq[i] = ||Y_i||^2
// ---------------------------------------------------------------------------
__global__ __launch_bounds__(256) void ysq_kernel(
    const float* __restrict__ F, const float* __restrict__ rinv,
    float* __restrict__ Y, float* __restrict__ sq) {
  const int i = blockIdx.x * 256 + threadIdx.x;
  const float ri = rinv[i];
  const float4* f4 = (const float4*)(F + (size_t)i * CC);
  float4* y4 = (float4*)(Y + (size_t)i * CC);
  float s = 0.0f;
#pragma unroll
  for (int c = 0; c < CC / 4; ++c) {
    float4 v = f4[c];
    v.x *= ri; v.y *= ri; v.z *= ri; v.w *= ri;
    y4[c] = v;
    s += v.x * v.x + v.y * v.y + v.z * v.z + v.w * v.w;
  }
  sq[i] = s;
}

// ---------------------------------------------------------------------------
// Fused IRLS partial step. Block (bx,by): waves handle row tiles
// I = bx*8 + wave, sweeping J tiles of slice by. Per 16x16 tile: transposed
// Gram via 16 f32 WMMAs, SCAD*Atilde weighting, C/D->B-operand layout fix
// (8 half-wave swaps), then 16 WMMAs accumulate OutT = F_J^T @ Wt^T and q.
// Y_J / F_J tiles staged to LDS with global_load_async_to_lds_b128.
// ---------------------------------------------------------------------------
__global__ __launch_bounds__(256) void irls_partial_kernel(
    const float* __restrict__ Adj, const float* __restrict__ Y,
    const float* __restrict__ sq, const float* __restrict__ rinv,
    const float* __restrict__ F, float* __restrict__ Pacc,
    float* __restrict__ Pq, const float* __restrict__ log_lams, int kstep) {
  __shared__ float Yl[16 * CC];
  __shared__ float Fl[16 * CC];
  __shared__ float sql[16];
  __shared__ float ril[16];

  const float lam = __expf(log_lams[kstep]);
  const int tid   = threadIdx.x;
  const int wave  = tid >> 5;
  const int lane  = tid & 31;
  const int li    = lane & 15;
  const int hf    = lane >> 4;
  const int I     = blockIdx.x * 8 + wave;  // row tile index
  const int irow  = I * 16 + li;            // this lane's output row i
  const int slice = blockIdx.y;
  const int Jlo   = slice * JPS;

  // Cached B operand for the Gram: Y_I^T (c x i)
  v2f bY[16];
  const float* Yi = Y + (size_t)irow * CC;
#pragma unroll
  for (int s = 0; s < 16; ++s) {
    bY[s].x = Yi[4 * s + 2 * hf];
    bY[s].y = Yi[4 * s + 2 * hf + 1];
  }
  const float sqi = sq[irow];
  const float ri  = rinv[irow];

  v8f outT[4];
#pragma unroll
  for (int cc = 0; cc < 4; ++cc)
    outT[cc] = (v8f){0.f,0.f,0.f,0.f,0.f,0.f,0.f,0.f};
  float qacc = 0.0f;

  const uint32_t lY = lds_off(Yl) + tid * 16;
  const uint32_t lF = lds_off(Fl) + tid * 16;

  for (int Jt = 0; Jt < JPS; ++Jt) {
    const int J = Jlo + Jt;
    __syncthreads();  // previous round's LDS reads complete
    {
      const uint64_t gY = (uint64_t)(Y + (size_t)J * 16 * CC) + tid * 16;
      const uint64_t gF = (uint64_t)(F + (size_t)J * 16 * CC) + tid * 16;
      asm volatile("global_load_async_to_lds_b128 %0, %1, off"
                   :: "v"(lY), "v"(gY) : "memory");
      asm volatile("global_load_async_to_lds_b128 %0, %1, off"
                   :: "v"(lF), "v"(gF) : "memory");
    }
    if (tid < 16) {
      sql[tid] = sq[J * 16 + tid];
      ril[tid] = rinv[J * 16 + tid];
    }
    asm volatile("s_wait_asynccnt 0x0" ::: "memory");
    __syncthreads();

    // Gram (transposed): g[r] -> G^T[jloc = r + 8*hf][i = li]
    v2f aY[16];
#pragma unroll
    for (int s = 0; s < 16; ++s) {
      aY[s].x = Yl[li * CC + 4 * s + 2 * hf];
      aY[s].y = Yl[li * CC + 4 * s + 2 * hf + 1];
    }
    v8f g = {0.f,0.f,0.f,0.f,0.f,0.f,0.f,0.f};
#pragma unroll
    for (int s = 0; s < 16; ++s)
      g = __builtin_amdgcn_wmma_f32_16x16x4_f32(false, aY[s], false, bY[s],
                                                (short)0, g, false, false);

    // Adjacency chunk: A[irow][J*16 + 8*hf + 0..7]
    const float4* Ar =
        (const float4*)(Adj + (size_t)irow * NN + J * 16 + 8 * hf);
    float4 a0 = Ar[0];
    float4 a1 = Ar[1];
    float av[8] = {a0.x, a0.y, a0.z, a0.w, a1.x, a1.y, a1.z, a1.w};

    float wt[8];
#pragma unroll
    for (int r = 0; r < 8; ++r) {
      const int jloc = r + 8 * hf;
      const int jrow = J * 16 + jloc;
      float z = sqi + sql[jloc] - 2.0f * g[r];
      z = fmaxf(z, 0.0f);
      const float y = sqrtf(z + kDistEps());
      const float mid =
          fmaxf(kScadA() * lam - y, 0.0f) * (1.0f / (kScadA() - 1.0f));
      const float rho = (y <= lam) ? lam : mid;
      const float w = fminf(rho / (2.0f * y), 1.0f);
      const float aval = av[r] + ((jrow == irow) ? 1.0f : 0.0f);
      const float wtv = w * aval * ri * ril[jloc];
      wt[r] = wtv;
      qacc += wtv;
    }

    // Convert Wt^T (C/D layout) -> B-operand layout
    float sw[8];
#pragma unroll
    for (int r = 0; r < 8; ++r) sw[r] = __shfl_xor(wt[r], 16, 32);
    v2f bw[4];
    bw[0].x = hf ? sw[2] : wt[0];  bw[0].y = hf ? sw[3] : wt[1];
    bw[1].x = hf ? sw[6] : wt[4];  bw[1].y = hf ? sw[7] : wt[5];
    bw[2].x = hf ? wt[2] : sw[0];  bw[2].y = hf ? wt[3] : sw[1];
    bw[3].x = hf ? wt[6] : sw[4];  bw[3].y = hf ? wt[7] : sw[5];

    // OutT[c][i] += F_J^T (c x j) @ Wt^T (j x i)
#pragma unroll
    for (int cc = 0; cc < 4; ++cc) {
      v2f aF[4];
#pragma unroll
      for (int gg = 0; gg < 4; ++gg) {
        aF[gg].x = Fl[(4 * gg + 2 * hf) * CC + cc * 16 + li];
        aF[gg].y = Fl[(4 * gg + 2 * hf + 1) * CC + cc * 16 + li];
      }
#pragma unroll
      for (int gg = 0; gg < 4; ++gg)
        outT[cc] = __builtin_amdgcn_wmma_f32_16x16x4_f32(
            false, aF[gg], false, bw[gg], (short)0, outT[cc], false, false);
    }
  }

  // Emit partials for this slice (no atomics; finalize kernel reduces)
  const float qtot = qacc + __shfl_xor(qacc, 16, 32);
  float* Pr = Pacc + ((size_t)slice * NN + irow) * CC;
#pragma unroll
  for (int cc = 0; cc < 4; ++cc)
#pragma unroll
    for (int r = 0; r < 8; ++r)
      Pr[cc * 16 + r + 8 * hf] = outT[cc][r];
  if (hf == 0) Pq[slice * NN + irow] = qtot;
}

// ---------------------------------------------------------------------------
// Reduce slice partials and apply (acc + lam*F0) / (q + lam)
// ---------------------------------------------------------------------------
__global__ __launch_bounds__(256) void finalize_kernel(
    const float* __restrict__ Pacc, const float* __restrict__ Pq,
    const float* __restrict__ lam_node, const float* __restrict__ F0,
    float* __restrict__ Fout) {
  const int t = blockIdx.x * 256 + threadIdx.x;  // 0 .. NN*CC-1
  const int row = t >> 6;
  const int c = t & (CC - 1);
  float acc = 0.0f, q = 0.0f;
#pragma unroll
  for (int s = 0; s < SLICES; ++s) {
    acc += Pacc[((size_t)s * NN + row) * CC + c];
    q   += Pq[s * NN + row];
  }
  const float lamn = lam_node[row];
  Fout[t] = (acc + lamn * F0[t]) / (q + lamn);
}

// ---------------------------------------------------------------------------
extern "C" void kernel_launch(void* const* d_in, const int* in_sizes, int n_in,
                              void* d_out, int out_size, void* d_ws,
                              size_t ws_size, hipStream_t stream) {
  const float* A        = (const float*)d_in[0];  // [4096,4096]
  const float* X        = (const float*)d_in[1];  // [4096,512]
  const float* W1       = (const float*)d_in[2];  // [512,256]
  const float* b1       = (const float*)d_in[3];  // [256]
  const float* W2       = (const float*)d_in[4];  // [256,64]
  const float* b2       = (const float*)d_in[5];  // [64]
  const float* log_lams = (const float*)d_in[6];  // [10]
  const float* raw_alpha= (const float*)d_in[7];  // [1]
  float* out = (float*)d_out;                     // [4096,64]

  float* ws       = (float*)d_ws;
  float* Hd       = ws;                       // 4096*256
  float* F0       = Hd + (size_t)NN * HH;     // 4096*64
  float* Fb       = F0 + (size_t)NN * CC;
  float* Fc       = Fb + (size_t)NN * CC;
  float* Yw       = Fc + (size_t)NN * CC;
  float* sq       = Yw + (size_t)NN * CC;     // 4096
  float* rinv     = sq + NN;
  float* raw_lam  = rinv + NN;
  float* lam_node = raw_lam + NN;
  float* accum    = lam_node + NN;            // 1
  float* Pacc     = accum + 1;                // SLICES*4096*64
  float* Pq       = Pacc + (size_t)SLICES * NN * CC;  // SLICES*4096

  init_kernel<<<1, 1, 0, stream>>>(accum);

  // MLP: Hd = relu(X@W1+b1); F0 = Hd@W2+b2
  gemm_bias_kernel<true><<<(NN / 16) * (HH / 16) / 8, 256, 0, stream>>>(
      X, W1, b1, Hd, NN, HH, DIN);
  gemm_bias_kernel<false><<<(NN / 16) * (CC / 16) / 8, 256, 0, stream>>>(
      Hd, W2, b2, F0, NN, CC, HH);

  conf_kernel<<<NN / 256, 256, 0, stream>>>(F0, raw_alpha, raw_lam, accum);
  lamnode_kernel<<<NN / 256, 256, 0, stream>>>(raw_lam, accum, lam_node);
  deg_kernel<<<NN, 256, 0, stream>>>(A, rinv);

  const float* Fin = F0;
  for (int k = 0; k < KSTEPS; ++k) {
    float* Fo = (k == KSTEPS - 1) ? out : ((k & 1) ? Fc : Fb);
    ysq_kernel<<<NN / 256, 256, 0, stream>>>(Fin, rinv, Yw, sq);
    dim3 grid(NN / 16 / 8, SLICES);
    irls_partial_kernel<<<grid, 256, 0, stream>>>(A, Yw, sq, rinv, Fin, Pacc,
                                                  Pq, log_lams, k);
    finalize_kernel<<<NN * CC / 256, 256, 0, stream>>>(Pacc, Pq, lam_node, F0,
                                                       Fo);
    Fin = Fo;
  }
}